// GIN_37873021616190
// MI455X (gfx1250) — compile-verified
//
#include <hip/hip_runtime.h>
#include <math.h>

typedef float v2f __attribute__((ext_vector_type(2)));
typedef float v8f __attribute__((ext_vector_type(8)));

#define N_NODES   100000
#define N_EDGES   1600000
#define N_FEAT    32
#define HID       16
#define N_CLASSES 10
#define N_GRAPHS  256
#define N_TILES   (N_NODES / 16)   // 6250, exact

// ---------------------------------------------------------------- zero scratch
__global__ void zero_ws_kernel(float* __restrict__ p, int n) {
    int i = blockIdx.x * blockDim.x + threadIdx.x;
    if (i < n) p[i] = 0.0f;
}

// ------------------------------------------------- edge scatter 1: agg1 += x[src]
// 8 threads per edge, each handles 4 contiguous features (float4 gather + 4 f32 atomics)
__global__ void scatter1_kernel(const float* __restrict__ x,
                                const int* __restrict__ src,
                                const int* __restrict__ dst,
                                float* __restrict__ agg) {
    int gid = blockIdx.x * blockDim.x + threadIdx.x;
    if (gid >= N_EDGES * 8) return;
    int e = gid >> 3;
    int q = (gid & 7) << 2;
    int s = src[e], d = dst[e];
    const float4 xv = *reinterpret_cast<const float4*>(x + (size_t)s * N_FEAT + q);
    float* ap = agg + (size_t)d * N_FEAT + q;
    unsafeAtomicAdd(ap + 0, xv.x);
    unsafeAtomicAdd(ap + 1, xv.y);
    unsafeAtomicAdd(ap + 2, xv.z);
    unsafeAtomicAdd(ap + 3, xv.w);
}

// ------------------------------------------------- GIN layer 1 via V_WMMA_F32_16X16X4_F32
// One wave per 16-node tile: (x+agg1)[16x32] @ W1[32x16] as 8 chained K=4 f32 WMMAs.
__global__ void gin1_kernel(const float* __restrict__ x,
                            const float* __restrict__ agg,
                            const float* __restrict__ W1,
                            const float* __restrict__ b1,
                            float* __restrict__ x1,
                            float* __restrict__ h) {
    const int lane = threadIdx.x & 31;
    const int wave = threadIdx.x >> 5;
    const int tile = blockIdx.x * 8 + wave;
    if (tile >= N_TILES) return;                 // wave-uniform: EXEC stays all-1s for WMMA
    const int m     = lane & 15;                 // A row / B col / D col
    const int khalf = (lane >> 4) << 1;          // lanes 0-15 -> K+{0,1}, lanes 16-31 -> K+{2,3}
    const int row   = tile * 16 + m;
    v8f acc = {};
#pragma unroll
    for (int c = 0; c < 8; ++c) {
        const int k = 4 * c + khalf;
        const float2 xa = *reinterpret_cast<const float2*>(x   + (size_t)row * N_FEAT + k);
        const float2 ga = *reinterpret_cast<const float2*>(agg + (size_t)row * N_FEAT + k);
        v2f a, b;
        a.x = xa.x + ga.x;  a.y = xa.y + ga.y;
        b.x = W1[(k + 0) * HID + m];
        b.y = W1[(k + 1) * HID + m];
        acc = __builtin_amdgcn_wmma_f32_16x16x4_f32(false, a, false, b,
                                                    (short)0, acc, false, false);
    }
    // D layout: VGPR v -> row v + 8*(lane>=16), col = lane&15
    const int   col   = m;
    const int   rbase = tile * 16 + ((lane >> 4) << 3);
    const float bias  = b1[col];
#pragma unroll
    for (int v = 0; v < 8; ++v) {
        const int r = rbase + v;
        float val = acc[v] + bias;
        x1[(size_t)r * HID + col] = val;
        h [(size_t)r * HID + col] = fmaxf(val, 0.0f);
    }
}

// ------------------------------------------------- edge scatter 2: agg2 += h[src]
__global__ void scatter2_kernel(const float* __restrict__ h,
                                const int* __restrict__ src,
                                const int* __restrict__ dst,
                                float* __restrict__ agg) {
    int gid = blockIdx.x * blockDim.x + threadIdx.x;
    if (gid >= N_EDGES * 4) return;
    int e = gid >> 2;
    int q = (gid & 3) << 2;
    int s = src[e], d = dst[e];
    const float4 hv = *reinterpret_cast<const float4*>(h + (size_t)s * HID + q);
    float* ap = agg + (size_t)d * HID + q;
    unsafeAtomicAdd(ap + 0, hv.x);
    unsafeAtomicAdd(ap + 1, hv.y);
    unsafeAtomicAdd(ap + 2, hv.z);
    unsafeAtomicAdd(ap + 3, hv.w);
}

// ------------------------------------------------- GIN layer 2 (WMMA) + graph pooling
// (h+agg2)[16x16] @ W2[16x10 zero-padded to 16] as 4 K=4 f32 WMMAs, then atomic
// scatter of x1 (cols 0-15) and x2 (cols 16-25) rows into gsum[graph][32].
__global__ void gin2_kernel(const float* __restrict__ h,
                            const float* __restrict__ agg2,
                            const float* __restrict__ W2,
                            const float* __restrict__ b2,
                            const float* __restrict__ x1,
                            const int* __restrict__ batch,
                            float* __restrict__ gsum) {
    const int lane = threadIdx.x & 31;
    const int wave = threadIdx.x >> 5;
    const int tile = blockIdx.x * 8 + wave;
    if (tile >= N_TILES) return;
    const int m     = lane & 15;
    const int khalf = (lane >> 4) << 1;
    const int row   = tile * 16 + m;
    v8f acc = {};
#pragma unroll
    for (int c = 0; c < 4; ++c) {
        const int k = 4 * c + khalf;
        const float2 ha = *reinterpret_cast<const float2*>(h    + (size_t)row * HID + k);
        const float2 ga = *reinterpret_cast<const float2*>(agg2 + (size_t)row * HID + k);
        v2f a, b;
        a.x = ha.x + ga.x;  a.y = ha.y + ga.y;
        b.x = (m < N_CLASSES) ? W2[(k + 0) * N_CLASSES + m] : 0.0f;
        b.y = (m < N_CLASSES) ? W2[(k + 1) * N_CLASSES + m] : 0.0f;
        acc = __builtin_amdgcn_wmma_f32_16x16x4_f32(false, a, false, b,
                                                    (short)0, acc, false, false);
    }
    const int   col   = m;
    const int   rbase = tile * 16 + ((lane >> 4) << 3);
    const float bias  = (col < N_CLASSES) ? b2[col] : 0.0f;
#pragma unroll
    for (int v = 0; v < 8; ++v) {
        const int r = rbase + v;
        const int g = batch[r];
        float* gp = gsum + (size_t)g * 32;
        unsafeAtomicAdd(gp + col, x1[(size_t)r * HID + col]);       // x1_s pooling
        if (col < N_CLASSES)
            unsafeAtomicAdd(gp + 16 + col, acc[v] + bias);          // x2_s pooling
    }
}

// ------------------------------------------------- log-softmax over [256, 26]
// One wave32 per graph; gsum padded stride 32, cols 0..25 map directly to output cols.
__global__ void softmax_kernel(const float* __restrict__ gsum, float* __restrict__ out) {
    const int lane = threadIdx.x & 31;
    const int wave = threadIdx.x >> 5;
    const int g = blockIdx.x * 8 + wave;
    if (g >= N_GRAPHS) return;
    const int  C     = HID + N_CLASSES;  // 26
    const bool valid = lane < C;
    float val = valid ? gsum[(size_t)g * 32 + lane] : -INFINITY;
    float mx = val;
#pragma unroll
    for (int o = 16; o > 0; o >>= 1) mx = fmaxf(mx, __shfl_xor(mx, o, 32));
    float e = valid ? __expf(val - mx) : 0.0f;
    float s = e;
#pragma unroll
    for (int o = 16; o > 0; o >>= 1) s += __shfl_xor(s, o, 32);
    if (valid) out[(size_t)g * C + lane] = val - mx - __logf(s);
}

// ---------------------------------------------------------------- launcher
extern "C" void kernel_launch(void* const* d_in, const int* in_sizes, int n_in,
                              void* d_out, int out_size, void* d_ws, size_t ws_size,
                              hipStream_t stream) {
    const float* x   = (const float*)d_in[0];
    const int*   ei  = (const int*)  d_in[1];   // [2, N_EDGES]: src row then dst row
    const int*   bat = (const int*)  d_in[2];
    const float* W1  = (const float*)d_in[3];
    const float* b1  = (const float*)d_in[4];
    const float* W2  = (const float*)d_in[5];
    const float* b2  = (const float*)d_in[6];
    float* out = (float*)d_out;

    // Workspace layout (floats): [agg1 | agg2 | gsum | x1 | h]
    float* ws   = (float*)d_ws;
    float* agg1 = ws;                                   // N*32
    float* agg2 = agg1 + (size_t)N_NODES * N_FEAT;      // N*16
    float* gsum = agg2 + (size_t)N_NODES * HID;         // 256*32
    float* x1   = gsum + (size_t)N_GRAPHS * 32;         // N*16
    float* h    = x1   + (size_t)N_NODES * HID;         // N*16

    const int nZero = N_NODES * N_FEAT + N_NODES * HID + N_GRAPHS * 32;
    zero_ws_kernel<<<(nZero + 255) / 256, 256, 0, stream>>>(ws, nZero);

    scatter1_kernel<<<(N_EDGES * 8) / 256, 256, 0, stream>>>(x, ei, ei + N_EDGES, agg1);
    gin1_kernel<<<(N_TILES + 7) / 8, 256, 0, stream>>>(x, agg1, W1, b1, x1, h);
    scatter2_kernel<<<(N_EDGES * 4) / 256, 256, 0, stream>>>(h, ei, ei + N_EDGES, agg2);
    gin2_kernel<<<(N_TILES + 7) / 8, 256, 0, stream>>>(h, agg2, W2, b2, x1, bat, gsum);
    softmax_kernel<<<N_GRAPHS / 8, 256, 0, stream>>>(gsum, out);
}